// GNNModel_16939351015990
// MI455X (gfx1250) — compile-verified
//
#include <hip/hip_runtime.h>
#include <hip/hip_bf16.h>
#include <math.h>

typedef float v2f __attribute__((ext_vector_type(2)));
typedef float v8f __attribute__((ext_vector_type(8)));

#define D 128   // hidden dim (H*C)
#define HH 4    // heads
#define CC 32   // channels per head

// ---------------------------------------------------------------------------
// WMMA f32 GEMM: out[M x Nout] = A[M x K] * W[K x Nout] + bias, optional lrelu.
// One wave (32 lanes) computes one 16x16 output tile with V_WMMA_F32_16X16X4_F32.
// Requires M % 16 == 0, Nout % 16 == 0, K % 4 == 0 (true for all our shapes).
// A-fragment (16x4, ISA 7.12.2): lanes 0-15 hold K={0,1}, lanes 16-31 K={2,3}.
// B-fragment (4x16): VGPR0 = K0/K2 rows, VGPR1 = K1/K3 rows, N striped on lanes.
// C/D: VGPR j -> row (j + 8*half), col = lane&15.
// ---------------------------------------------------------------------------
__global__ void wmma_gemm_f32(const float* __restrict__ A,
                              const float* __restrict__ W,
                              const float* __restrict__ bias,
                              float* __restrict__ out,
                              int M, int K, int Nout, int act) {
    const int mt   = blockIdx.x * 16;
    const int nt   = blockIdx.y * 16;
    const int lane = threadIdx.x & 31;
    const int half = lane >> 4;     // 0: K 0,1 ; 1: K 2,3
    const int lr   = lane & 15;

    v8f acc = {};
    const float* arow = A + (size_t)(mt + lr) * K;
    const int k0 = half * 2;

    for (int k = 0; k < K; k += 4) {
        v2f a, b;
        a.x = arow[k + k0];
        a.y = arow[k + k0 + 1];
        b.x = W[(size_t)(k + k0)     * Nout + nt + lr];
        b.y = W[(size_t)(k + k0 + 1) * Nout + nt + lr];
        acc = __builtin_amdgcn_wmma_f32_16x16x4_f32(
            /*neg_a=*/false, a, /*neg_b=*/false, b,
            /*c_mod=*/(short)0, acc, /*reuse_a=*/false, /*reuse_b=*/false);
    }

    const float bv = bias ? bias[nt + lr] : 0.0f;
#pragma unroll
    for (int j = 0; j < 8; ++j) {
        int   row = mt + j + half * 8;
        float val = acc[j] + bv;
        if (act) val = val > 0.0f ? val : 0.01f * val;
        out[(size_t)row * Nout + nt + lr] = val;
    }
}

// monotone float<->uint encoding for atomicMax on floats
__device__ __forceinline__ unsigned enc_f(float f) {
    unsigned u = __float_as_uint(f);
    return u ^ ((u >> 31) ? 0xFFFFFFFFu : 0x80000000u);
}
__device__ __forceinline__ float dec_f(unsigned u) {
    return (u >> 31) ? __uint_as_float(u ^ 0x80000000u) : __uint_as_float(~u);
}

// ---------------------------------------------------------------------------
// Edge pass A: wave per edge. alpha[e,h] = <q[dst,h,:], k[src,h,:]+e[h,:]>/sqrt(C)
// and atomicMax segment max per (dst, h).
// ---------------------------------------------------------------------------
__global__ void edge_alpha_max(const float* __restrict__ q,
                               const float* __restrict__ kk,
                               const int* __restrict__ src,
                               const int* __restrict__ dst,
                               const float* __restrict__ ea,
                               const float* __restrict__ We,   // [2 x 128]
                               float* __restrict__ alpha,      // [E x 4]
                               unsigned* __restrict__ menc,    // [N x 4]
                               int E) {
    const int e    = (int)((blockIdx.x * blockDim.x + threadIdx.x) >> 5);
    const int lane = threadIdx.x & 31;
    if (e >= E) return;

    const int   s  = src[e], d = dst[e];
    const float a0 = ea[(size_t)e * 2 + 0];
    const float a1 = ea[(size_t)e * 2 + 1];
    const float* qrow = q  + (size_t)d * D;
    const float* krow = kk + (size_t)s * D;

#pragma unroll
    for (int h = 0; h < HH; ++h) {
        const int   c  = h * CC + lane;
        const float ev = a0 * We[c] + a1 * We[D + c];
        float p = qrow[c] * (krow[c] + ev);
#pragma unroll
        for (int off = 16; off; off >>= 1) p += __shfl_xor(p, off, 32);
        if (lane == 0) {
            const float al = p * 0.17677669529663689f;   // 1/sqrt(32)
            alpha[(size_t)e * HH + h] = al;
            atomicMax(&menc[(size_t)d * HH + h], enc_f(al));
        }
    }
}

// ---------------------------------------------------------------------------
// Edge pass B: wave per edge. p = exp(alpha - m[dst]); denom[dst,h] += p;
// agg[dst,h,:] += p * (v[src,h,:] + e[h,:]). Normalization deferred to node pass.
// ---------------------------------------------------------------------------
__global__ void edge_scatter(const float* __restrict__ v,
                             const float* __restrict__ alpha,
                             const unsigned* __restrict__ menc,
                             const int* __restrict__ src,
                             const int* __restrict__ dst,
                             const float* __restrict__ ea,
                             const float* __restrict__ We,
                             float* __restrict__ denom,   // [N x 4]
                             float* __restrict__ agg,     // [N x 128]
                             int E) {
    const int e    = (int)((blockIdx.x * blockDim.x + threadIdx.x) >> 5);
    const int lane = threadIdx.x & 31;
    if (e >= E) return;

    const int   s  = src[e], d = dst[e];
    const float a0 = ea[(size_t)e * 2 + 0];
    const float a1 = ea[(size_t)e * 2 + 1];
    const float* vrow = v + (size_t)s * D;

#pragma unroll
    for (int h = 0; h < HH; ++h) {
        const float m  = dec_f(menc[(size_t)d * HH + h]);
        const float p  = __expf(alpha[(size_t)e * HH + h] - m);
        if (lane == 0) atomicAdd(&denom[(size_t)d * HH + h], p);
        const int   c  = h * CC + lane;
        const float ev = a0 * We[c] + a1 * We[D + c];
        atomicAdd(&agg[(size_t)d * D + c], p * (vrow[c] + ev));
    }
}

// ---------------------------------------------------------------------------
// Node update: out = lrelu(agg/(denom+1e-16) + skip)
// ---------------------------------------------------------------------------
__global__ void node_update(const float* __restrict__ agg,
                            const float* __restrict__ denom,
                            const float* __restrict__ skip,
                            float* __restrict__ out, int total) {
    const int i = blockIdx.x * blockDim.x + threadIdx.x;
    if (i >= total) return;
    const int n = i >> 7;           // /128
    const int c = i & (D - 1);
    const int h = c >> 5;           // /32
    float val = agg[i] / (denom[(size_t)n * HH + h] + 1e-16f) + skip[i];
    out[i] = val > 0.0f ? val : 0.01f * val;
}

// ---------------------------------------------------------------------------
// Head: logits = emb[32] @ W2[32x8] + b2, then log_softmax over 8.
// ---------------------------------------------------------------------------
__global__ void head_logsoftmax(const float* __restrict__ emb,
                                const float* __restrict__ W2,
                                const float* __restrict__ b2,
                                float* __restrict__ out, int N) {
    const int n = blockIdx.x * blockDim.x + threadIdx.x;
    if (n >= N) return;
    float logits[8];
#pragma unroll
    for (int j = 0; j < 8; ++j) logits[j] = b2[j];
    const float* er = emb + (size_t)n * CC;
#pragma unroll 4
    for (int c = 0; c < CC; ++c) {
        const float ev = er[c];
#pragma unroll
        for (int j = 0; j < 8; ++j) logits[j] += ev * W2[c * 8 + j];
    }
    float mx = logits[0];
#pragma unroll
    for (int j = 1; j < 8; ++j) mx = fmaxf(mx, logits[j]);
    float sum = 0.0f;
#pragma unroll
    for (int j = 0; j < 8; ++j) sum += __expf(logits[j] - mx);
    const float lse = mx + __logf(sum);
#pragma unroll
    for (int j = 0; j < 8; ++j) out[(size_t)n * 8 + j] = logits[j] - lse;
}

// ---------------------------------------------------------------------------
static inline float* ws_take(char** p, size_t floats) {
    float* r = (float*)*p;
    *p += ((floats * sizeof(float)) + 255) & ~(size_t)255;
    return r;
}

extern "C" void kernel_launch(void* const* d_in, const int* in_sizes, int n_in,
                              void* d_out, int out_size, void* d_ws, size_t ws_size,
                              hipStream_t stream) {
    const float* x    = (const float*)d_in[0];
    const int*   eidx = (const int*)  d_in[1];
    const float* ea   = (const float*)d_in[2];
    const float* Wq   = (const float*)d_in[3];
    const float* bq   = (const float*)d_in[4];
    const float* Wk   = (const float*)d_in[5];
    const float* bk   = (const float*)d_in[6];
    const float* Wv   = (const float*)d_in[7];
    const float* bv   = (const float*)d_in[8];
    const float* We   = (const float*)d_in[9];
    const float* Ws   = (const float*)d_in[10];
    const float* bs   = (const float*)d_in[11];
    const float* W1   = (const float*)d_in[12];
    const float* b1   = (const float*)d_in[13];
    const float* W2   = (const float*)d_in[14];
    const float* b2   = (const float*)d_in[15];

    const int N = in_sizes[0] / D;       // 50000
    const int E = in_sizes[1] / 2;       // 400000
    const int* src = eidx;
    const int* dst = eidx + E;

    // workspace carve-out (~190 MB, all fits in d_ws; whole working set < 192MB L2)
    char* p = (char*)d_ws;
    float* hA    = ws_take(&p, (size_t)N * D);
    float* hB    = ws_take(&p, (size_t)N * D);
    float* qbuf  = ws_take(&p, (size_t)N * D);
    float* kbuf  = ws_take(&p, (size_t)N * D);
    float* vbuf  = ws_take(&p, (size_t)N * D);
    float* sbuf  = ws_take(&p, (size_t)N * D);
    float* agg   = ws_take(&p, (size_t)N * D);
    float* alpha = ws_take(&p, (size_t)E * HH);
    float* denom = ws_take(&p, (size_t)N * HH);
    unsigned* menc = (unsigned*)ws_take(&p, (size_t)N * HH);

    const dim3 gemmGrid(N / 16, D / 16);     // 3125 x 8 tiles
    const dim3 gemmBlk(32);                  // one wave per tile (EXEC all-1s)
    const int  edgeBlocks = (E + 7) / 8;     // 8 waves of 32 per 256-thread block
    const int  nodeBlocks = (N * D + 255) / 256;

    const float* hin = x;
    for (int l = 0; l < 3; ++l) {
        const size_t wOff = (size_t)l * D * D;
        const size_t bOff = (size_t)l * D;
        const float* Wel  = We + (size_t)l * 2 * D;
        float* hout = (l & 1) ? hB : hA;     // l0->hA, l1->hB, l2->hA

        wmma_gemm_f32<<<gemmGrid, gemmBlk, 0, stream>>>(hin, Wq + wOff, bq + bOff, qbuf, N, D, D, 0);
        wmma_gemm_f32<<<gemmGrid, gemmBlk, 0, stream>>>(hin, Wk + wOff, bk + bOff, kbuf, N, D, D, 0);
        wmma_gemm_f32<<<gemmGrid, gemmBlk, 0, stream>>>(hin, Wv + wOff, bv + bOff, vbuf, N, D, D, 0);
        wmma_gemm_f32<<<gemmGrid, gemmBlk, 0, stream>>>(hin, Ws + wOff, bs + bOff, sbuf, N, D, D, 0);

        hipMemsetAsync(menc,  0, (size_t)N * HH * sizeof(unsigned), stream);
        hipMemsetAsync(denom, 0, (size_t)N * HH * sizeof(float),    stream);
        hipMemsetAsync(agg,   0, (size_t)N * D  * sizeof(float),    stream);

        edge_alpha_max<<<edgeBlocks, 256, 0, stream>>>(qbuf, kbuf, src, dst, ea, Wel,
                                                       alpha, menc, E);
        edge_scatter<<<edgeBlocks, 256, 0, stream>>>(vbuf, alpha, menc, src, dst, ea, Wel,
                                                     denom, agg, E);
        node_update<<<nodeBlocks, 256, 0, stream>>>(agg, denom, sbuf, hout, N * D);
        hin = hout;
    }

    // emb = lrelu(h @ W1 + b1): [N,128]x[128,32], reuse qbuf
    const dim3 embGrid(N / 16, CC / 16);
    wmma_gemm_f32<<<embGrid, gemmBlk, 0, stream>>>(hin, W1, b1, qbuf, N, D, CC, 1);

    head_logsoftmax<<<(N + 255) / 256, 256, 0, stream>>>(qbuf, W2, b2, (float*)d_out, N);
}